// ScaledDotProductAttention_1_44272522887185
// MI455X (gfx1250) — compile-verified
//
#include <hip/hip_runtime.h>

typedef float v2f __attribute__((ext_vector_type(2)));
typedef float v4f __attribute__((ext_vector_type(4)));
typedef float v8f __attribute__((ext_vector_type(8)));

#define B_    4
#define C_    5
#define H_    200
#define DK_   20
#define DIN_  400
#define BCH_  (B_ * C_ * H_)     // 4000
#define HT_   208                // 13 tiles of 16
#define SSTR_ 212                // LDS stride for s_row (bank-friendly)

// ---------------------------------------------------------------------------
// Phase 1: projections q/k/v (contract n=400) and c (contract n=20).
// One block per (b,c,h) slice; blockIdx.y selects {Q,K,V,cdd}.
// 320 threads: thread = (row-group rr 0..63, chunk c4 0..4). Each thread does
// a 16B global_load_b128 of 4 consecutive d-values of one row -> v4f FMA with
// W[n]; rows strided by 64. Then a 64->1 LDS tree reduction over rows.
// ---------------------------------------------------------------------------
__global__ __launch_bounds__(320)
void proj_kernel(const float* __restrict__ Q, const float* __restrict__ K,
                 const float* __restrict__ V, const float* __restrict__ cdd,
                 const float* __restrict__ W, const float* __restrict__ bptr,
                 const float* __restrict__ W1, const float* __restrict__ b1ptr,
                 float* __restrict__ ws)
{
    const int bch   = blockIdx.x;     // 0..3999
    const int which = blockIdx.y;     // 0=Q 1=K 2=V 3=cdd
    const int tid   = threadIdx.x;    // 0..319
    const int c4    = tid % 5;        // 4-float chunk within a 20-float row
    const int rr    = tid / 5;        // 0..63 row group

    __shared__ float part[64 * DK_];  // [row-group][d]

    v4f acc = {0.f, 0.f, 0.f, 0.f};
    if (which < 3) {
        const float* X = (which == 0) ? Q : (which == 1) ? K : V;
        const float* base = X + (size_t)bch * (DIN_ * DK_);
        #pragma unroll
        for (int j = 0; j < 6; ++j) {              // rows 0..383
            const int n = rr + 64 * j;
            const v4f x = *(const v4f*)(base + n * DK_ + 4 * c4);
            const float w = W[n];
            acc += x * w;
        }
        {                                           // rows 384..399 (rr < 16)
            const int n  = 384 + (rr & 15);
            const v4f x  = *(const v4f*)(base + n * DK_ + 4 * c4);
            const float w = (rr < 16) ? W[n] : 0.f;
            acc += x * w;
        }
    } else {
        const float* base = cdd + (size_t)bch * (DK_ * DK_);
        const int n   = (rr < DK_) ? rr : (DK_ - 1);    // clamp, zero weight
        const v4f x   = *(const v4f*)(base + n * DK_ + 4 * c4);
        const float w = (rr < DK_) ? W1[n] : 0.f;
        acc = x * w;
    }
    *(v4f*)(part + rr * DK_ + 4 * c4) = acc;
    __syncthreads();

    // 64 -> 1 tree reduction over row groups
    #pragma unroll
    for (int s = 32; s >= 1; s >>= 1) {
        for (int i = tid; i < s * DK_; i += 320)
            part[i] += part[i + s * DK_];
        __syncthreads();
    }
    if (tid < DK_) {
        const float bias = (which < 3) ? *bptr : *b1ptr;
        ws[(size_t)which * (BCH_ * DK_) + (size_t)bch * DK_ + tid] = part[tid] + bias;
    }
}

// ---------------------------------------------------------------------------
// Phase 2: scores = (q·k^T + k·c^T)/sqrt(dk); exp; normalize; context = attn·v
// Grid (20 bc-pairs, 13 m-tiles) x 128 threads (4 waves).
// fp32 WMMA 16x16x4:
//   A (16x4): lane L supplies A[L%16][2*(L/16)+j] in vgpr j
//   B (4x16): lane L supplies B[2*(L/16)+j][L%16]
//   C/D    : vgpr j, lane L = D[j + 8*(L/16)][L%16]
// All operand fetches use clamped indices + v_cndmask (no exec-divergent
// loads), keeping EXEC all-1s around the WMMAs.
// ---------------------------------------------------------------------------
__global__ __launch_bounds__(128)
void attn_kernel(const float* __restrict__ ws, const int* __restrict__ dkp,
                 float* __restrict__ ctx_out, float* __restrict__ attn_out)
{
    const float* qp = ws;
    const float* kp = ws + 1 * BCH_ * DK_;
    const float* vp = ws + 2 * BCH_ * DK_;
    const float* cp = ws + 3 * BCH_ * DK_;

    const int bc   = blockIdx.x;          // 0..19
    const int mt   = blockIdx.y;          // 0..12
    const int tid  = threadIdx.x;
    const int wave = tid >> 5;
    const int lane = tid & 31;
    const int l16  = lane & 15;
    const int kh   = lane >> 4;           // K-half select

    const float* qb = qp + (size_t)bc * H_ * DK_;
    const float* kb = kp + (size_t)bc * H_ * DK_;
    const float* vb = vp + (size_t)bc * H_ * DK_;
    const float* cb = cp + (size_t)bc * H_ * DK_;

    __shared__ float s_row[16 * SSTR_];   // exp'ed (then normalized) score strip
    __shared__ float rdenom[16];

    const float scale = rsqrtf((float)(*dkp));
    const int   mrow  = mt * 16 + l16;
    const int   mrc   = (mrow < H_) ? mrow : (H_ - 1);   // clamped
    const bool  mok   = (mrow < H_);
    const v2f   z2    = {0.f, 0.f};

    // ---- score tiles: S = q·k^T + k·c^T, then exp, into LDS ----
    for (int nt = wave; nt < 13; nt += 4) {
        const int  nrow = nt * 16 + l16;
        const int  nrc  = (nrow < H_) ? nrow : (H_ - 1);
        const bool nok  = (nrow < H_);
        v8f acc = {};
        #pragma unroll
        for (int k0 = 0; k0 < DK_; k0 += 4) {
            const int ka = k0 + 2 * kh;               // even -> 8B aligned
            v2f a = *(const v2f*)(qb + mrc * DK_ + ka);
            v2f b = *(const v2f*)(kb + nrc * DK_ + ka);
            a = mok ? a : z2;
            b = nok ? b : z2;
            acc = __builtin_amdgcn_wmma_f32_16x16x4_f32(false, a, false, b,
                                                        (short)0, acc, false, false);
            v2f a2 = *(const v2f*)(kb + mrc * DK_ + ka);
            v2f b2 = *(const v2f*)(cb + nrc * DK_ + ka);
            a2 = mok ? a2 : z2;
            b2 = nok ? b2 : z2;
            acc = __builtin_amdgcn_wmma_f32_16x16x4_f32(false, a2, false, b2,
                                                        (short)0, acc, false, false);
        }
        const int g = nt * 16 + l16;
        #pragma unroll
        for (int j = 0; j < 8; ++j) {
            const int ml = j + 8 * kh;
            s_row[ml * SSTR_ + g] = (g < H_) ? __expf(acc[j] * scale) : 0.f;
        }
    }
    __syncthreads();

    // ---- row sums -> reciprocal denominators ----
    if (tid < 16) {
        float s = 0.f;
        for (int g = 0; g < H_; ++g) s += s_row[tid * SSTR_ + g];
        rdenom[tid] = 1.f / (s + 1e-8f);
    }
    __syncthreads();

    // ---- normalize strip in place ----
    for (int i = tid; i < 16 * HT_; i += 128) {
        const int ml = i / HT_, g = i % HT_;
        s_row[ml * SSTR_ + g] *= rdenom[ml];
    }
    __syncthreads();

    // ---- write attn ----
    for (int i = tid; i < 16 * H_; i += 128) {
        const int ml = i / H_, g = i % H_;
        const int m  = mt * 16 + ml;
        if (m < H_)
            attn_out[(size_t)(bc * H_ + m) * H_ + g] = s_row[ml * SSTR_ + g];
    }

    // ---- context = attn · v via WMMA, waves 0/1 take d-tiles 0..15 / 16..31 ----
    if (wave < 2) {
        const int  dcol = wave * 16 + l16;
        const int  dc   = (dcol < DK_) ? dcol : (DK_ - 1);
        const bool dok  = (dcol < DK_);
        v8f acc = {};
        for (int k0 = 0; k0 < HT_; k0 += 4) {
            const int ka  = k0 + 2 * kh;
            const int k0c = (ka     < H_) ? ka     : (H_ - 1);
            const int k1c = (ka + 1 < H_) ? ka + 1 : (H_ - 1);
            v2f a;
            a.x = s_row[l16 * SSTR_ + ka];
            a.y = s_row[l16 * SSTR_ + ka + 1];
            float bx = vb[k0c * DK_ + dc];
            float by = vb[k1c * DK_ + dc];
            v2f b;
            b.x = (dok && ka     < H_) ? bx : 0.f;
            b.y = (dok && ka + 1 < H_) ? by : 0.f;
            acc = __builtin_amdgcn_wmma_f32_16x16x4_f32(false, a, false, b,
                                                        (short)0, acc, false, false);
        }
        #pragma unroll
        for (int j = 0; j < 8; ++j) {
            const int m = mt * 16 + j + 8 * kh;
            if (m < H_ && dok)
                ctx_out[(size_t)(bc * H_ + m) * DK_ + dcol] = acc[j];
        }
    }
}

// ---------------------------------------------------------------------------
extern "C" void kernel_launch(void* const* d_in, const int* in_sizes, int n_in,
                              void* d_out, int out_size, void* d_ws, size_t ws_size,
                              hipStream_t stream)
{
    const float* Q   = (const float*)d_in[0];
    const float* K   = (const float*)d_in[1];
    const float* V   = (const float*)d_in[2];
    const float* cdd = (const float*)d_in[3];
    const float* W   = (const float*)d_in[4];
    const float* b   = (const float*)d_in[5];
    const float* W1  = (const float*)d_in[6];
    const float* b1  = (const float*)d_in[7];
    const int*   dk  = (const int*)d_in[8];

    float* ws       = (float*)d_ws;                 // 4 * 80000 floats = 1.28 MB
    float* ctx_out  = (float*)d_out;                // 80000 f32
    float* attn_out = (float*)d_out + BCH_ * DK_;   // 800000 f32

    proj_kernel<<<dim3(BCH_, 4), 320, 0, stream>>>(Q, K, V, cdd, W, b, W1, b1, ws);
    attn_kernel<<<dim3(B_ * C_, 13), 128, 0, stream>>>(ws, dk, ctx_out, attn_out);
}